// Spline_12919261626950
// MI455X (gfx1250) — compile-verified
//
#include <hip/hip_runtime.h>
#include <math.h>

// Rational-quadratic spline flow (forward + logdet), MI455X / gfx1250.
// Streaming fp32: ~201 MB HBM traffic -> ~8.6 us floor at 23.3 TB/s; body is
// ~50 issue slots/element so the kernel sits near the compute/BW crossover.
// Per-feature tables rebuilt per block into LDS (native transcendentals, off
// the critical path). Hot loop: v_med3 clamp + 3-level branchless binary
// search over register knots + 2x ds_load_b128 gather + fused rational eval
// with v_rcp_f32(+NR) and one v_log_f32. Unrolled x2 for MLP/dual-issue.

__global__ __launch_bounds__(256) void spline_fwd_kernel(
    const float* __restrict__ x,
    const float* __restrict__ width,
    const float* __restrict__ height,
    const float* __restrict__ derivative,
    float* __restrict__ out,
    float* __restrict__ logdet,
    int n)
{
    constexpr int D = 64;
    constexpr int K = 8;          // N_BINS
    constexpr float MINW = 1e-3f; // MIN_BIN_WIDTH
    constexpr float MINH = 1e-3f; // MIN_BIN_HEIGHT
    constexpr float MIND = 1e-3f; // MIN_DERIVATIVE

    // Stride 9 (odd) per feature to break LDS bank aliasing for gathers.
    __shared__ float4 sA[D * 9];    // {cw_k, 1/w_k, ch_k, h_k}
    __shared__ float4 sB[D * 9];    // {delta_k, d_k, d_{k+1}, 0}
    __shared__ float  sScan[D * 9]; // interior knots cw_{k+1}, k = 0..6

    const int tid = threadIdx.x;

    // ---- Per-block table precompute: one feature per thread (tid < 64) ----
    if (tid < D) {
        const int d = tid;

        // softmax(width row) -> min-size bins -> cumulative knots in [0,1]
        float wv[K];
        #pragma unroll
        for (int k = 0; k < K; ++k) wv[k] = width[d * K + k];
        float mw = wv[0];
        #pragma unroll
        for (int k = 1; k < K; ++k) mw = fmaxf(mw, wv[k]);
        float ew[K]; float sw = 0.f;
        #pragma unroll
        for (int k = 0; k < K; ++k) { ew[k] = __expf(wv[k] - mw); sw += ew[k]; }
        float invsw = 1.0f / sw;
        float cw[K + 1]; cw[0] = 0.f;
        {
            float c = 0.f;
            #pragma unroll
            for (int k = 0; k < K; ++k) {
                float p  = ew[k] * invsw;
                c += MINW + (1.0f - MINW * (float)K) * p;
                cw[k + 1] = c;
            }
        }
        cw[K] = 1.0f;  // force exact upper knot

        // same for heights
        float hv[K];
        #pragma unroll
        for (int k = 0; k < K; ++k) hv[k] = height[d * K + k];
        float mh = hv[0];
        #pragma unroll
        for (int k = 1; k < K; ++k) mh = fmaxf(mh, hv[k]);
        float eh[K]; float sh = 0.f;
        #pragma unroll
        for (int k = 0; k < K; ++k) { eh[k] = __expf(hv[k] - mh); sh += eh[k]; }
        float invsh = 1.0f / sh;
        float ch[K + 1]; ch[0] = 0.f;
        {
            float c = 0.f;
            #pragma unroll
            for (int k = 0; k < K; ++k) {
                float p  = eh[k] * invsh;
                c += MINH + (1.0f - MINH * (float)K) * p;
                ch[k + 1] = c;
            }
        }
        ch[K] = 1.0f;

        // derivatives: 1e-3 + softplus(v); |v| <= 1e-3 so 1+e^v ~= 2 (safe)
        float dv[K + 1];
        #pragma unroll
        for (int k = 0; k <= K; ++k) {
            float v = derivative[d * (K + 1) + k];
            dv[k] = MIND + __logf(1.0f + __expf(v));
        }

        #pragma unroll
        for (int k = 0; k < K; ++k) {
            float w  = cw[k + 1] - cw[k];
            float h  = ch[k + 1] - ch[k];
            float iw = 1.0f / w;   // precise div: precompute only
            float dl = h / w;      // delta = heights/widths (matches reference)
            sA[d * 9 + k] = make_float4(cw[k], iw, ch[k], h);
            sB[d * 9 + k] = make_float4(dl, dv[k], dv[k + 1], 0.0f);
            sScan[d * 9 + k] = cw[k + 1];
        }
    }
    __syncthreads();

    const int gtid   = blockIdx.x * blockDim.x + tid;
    const int stride = gridDim.x * blockDim.x;   // multiple of 64 (blockDim=256)
    const int d      = tid & (D - 1);            // == element % 64, fixed per thread

    // Interior knots cw[1..7] in registers (sorted): idx = #knots <= xc in 0..7.
    // (knot cw[8]==1.0 and the clip(...,7) cancel, so 7 knots suffice.)
    float rcw[K - 1];
    #pragma unroll
    for (int k = 0; k < K - 1; ++k) rcw[k] = sScan[d * 9 + k];
    const int dbase = d * 9;

    auto body = [&](int e) {
        float xv = __builtin_nontemporal_load(x + e);
        float xc = __builtin_amdgcn_fmed3f(xv, 0.0f, 1.0f);  // v_med3 clamp

        // 3-level branchless binary search: idx = count(xc >= cw[k], k=1..7)
        bool  b2  = xc >= rcw[3];              // idx >= 4  (knot cw[4])
        float mB  = b2 ? rcw[5] : rcw[1];      // cw[6] : cw[2]
        bool  b1  = xc >= mB;
        float mLo = b1 ? rcw[2] : rcw[0];      // cw[3] : cw[1]
        float mHi = b1 ? rcw[6] : rcw[4];      // cw[7] : cw[5]
        float mC  = b2 ? mHi : mLo;
        bool  b0  = xc >= mC;
        int idx = (b2 ? 4 : 0) | (b1 ? 2 : 0) | (b0 ? 1 : 0);

        float4 a = sA[dbase + idx];   // ds_load_b128
        float4 b = sB[dbase + idx];   // ds_load_b128

        float theta = (xc - a.x) * a.y;
        float omt   = 1.0f - theta;
        float t1mt  = theta * omt;
        float th2   = theta * theta;

        float delta = b.x, dk = b.y, dk1 = b.z;

        float denom = fmaf((dk1 + dk) - 2.0f * delta, t1mt, delta);
        float numer = a.w * fmaf(delta, th2, dk * t1mt);

        // fast reciprocal + one Newton step (~0.5 ulp); denom in [~0.69, ~1]
        float r = __builtin_amdgcn_rcpf(denom);
        r = r * fmaf(-denom, r, 2.0f);

        float y = fmaf(numer, r, a.z);

        float inner = fmaf(dk1, th2, fmaf(2.0f * delta, t1mt, dk * (omt * omt)));
        float dnum  = (delta * delta) * inner;

        // log(dnum) - 2*log(denom) == log(dnum * r * r): one v_log_f32
        float lad = __logf(dnum * r * r);

        // finite inputs: inside [0,1] <=> clamp is identity (one compare)
        bool inside = (xv == xc);
        __builtin_nontemporal_store(inside ? y   : xv,   out    + e);
        __builtin_nontemporal_store(inside ? lad : 0.0f, logdet + e);
    };

    // x2 unrolled grid-stride: both loads + both gather pairs in flight.
    int e = gtid;
    for (; e + stride < n; e += 2 * stride) {
        // speculative prefetch of the next unrolled iteration (global_prefetch_b8)
        __builtin_prefetch(x + e + 2 * stride, 0, 0);
        __builtin_prefetch(x + e + 3 * stride, 0, 0);
        body(e);
        body(e + stride);
    }
    if (e < n) body(e);
}

extern "C" void kernel_launch(void* const* d_in, const int* in_sizes, int n_in,
                              void* d_out, int out_size, void* d_ws, size_t ws_size,
                              hipStream_t stream) {
    const float* x          = (const float*)d_in[0];
    const float* width      = (const float*)d_in[1];
    const float* height     = (const float*)d_in[2];
    const float* derivative = (const float*)d_in[3];
    float* out = (float*)d_out;

    const int n = in_sizes[0];          // B*D = 16,777,216
    float* logdet = out + n;            // outputs concatenated: [y | logdet]

    const int threads = 256;
    int need = (n + threads - 1) / threads;
    int blocks = need < 2048 ? need : 2048;   // ~16 unrolled iters/thread
    if (blocks < 1) blocks = 1;

    spline_fwd_kernel<<<dim3(blocks), dim3(threads), 0, stream>>>(
        x, width, height, derivative, out, logdet, n);
}